// LiquidLoRALayer_70145405878766
// MI455X (gfx1250) — compile-verified
//
#include <hip/hip_runtime.h>

typedef __attribute__((ext_vector_type(2))) float v2f;
typedef __attribute__((ext_vector_type(8))) float v8f;

#define R 64
#define LDSP 68   // 64 + 4 pad: 16 lanes * stride 68 floats -> banks 4L..4L+3, all 64 banks once

// ---------------------------------------------------------------------------
// Kernel 1: liquid dynamics. 3 closed-form integration steps on h[O,R].
// 256 threads = 4 rows x 64 lanes-per-row; inp (=[target,h]) shared via LDS.
// ---------------------------------------------------------------------------
__global__ __launch_bounds__(256) void liquid_kernel(
    const float* __restrict__ loraB, const float* __restrict__ hidden0,
    const float* __restrict__ Wg, const float* __restrict__ bg,
    const float* __restrict__ Wt, const float* __restrict__ bt,
    float* __restrict__ Beff)
{
    __shared__ float inp[4][2 * R];
    const int rl = threadIdx.x >> 6;     // row within block 0..3
    const int j  = threadIdx.x & 63;     // 0..63
    const int o  = blockIdx.x * 4 + rl;

    const float tgt = loraB[(size_t)o * R + j];
    float h = hidden0[(size_t)o * R + j];
    const float bgj = bg[j], btj = bt[j];
    inp[rl][j] = tgt;

    for (int step = 0; step < 3; ++step) {
        inp[rl][R + j] = h;
        __syncthreads();
        float ag = 0.f, at = 0.f;
        #pragma unroll 8
        for (int k = 0; k < 2 * R; ++k) {
            const float v = inp[rl][k];
            ag = __builtin_fmaf(v, Wg[j * (2 * R) + k], ag);
            at = __builtin_fmaf(v, Wt[j * (2 * R) + k], at);
        }
        const float f   = 1.0f / (1.0f + expf(-(ag + bgj)));
        const float sg  = 1.0f / (1.0f + expf(-(at + btj)));
        const float tau = 0.1f + 9.9f * sg;
        const float a   = 1.0f / tau + f;
        const float dec = expf(-a * 0.1f);
        h = h * dec + (f / a) * tgt * (1.0f - dec);
        __syncthreads();
    }
    Beff[(size_t)o * R + j] = h;
}

// ---------------------------------------------------------------------------
// Kernel 2: t[M,64] = x[M,D] @ lora_A[64,D]^T   (f32 WMMA 16x16x4)
// 128 threads = 4 waves; block M-tile = 64, full N = 64; K staged by 64.
// Software-pipelined: global loads of chunk k0+64 issue before the WMMA loop
// over chunk k0, so VMEM latency overlaps compute (critical at ~1 wave/SIMD).
// ---------------------------------------------------------------------------
__global__ __launch_bounds__(128) void gemm1_kernel(
    const float* __restrict__ x, const float* __restrict__ A,
    float* __restrict__ t, int D)
{
    __shared__ float xs[64][LDSP];
    __shared__ float as[64][LDSP];

    const int m0   = blockIdx.x * 64;
    const int tid  = threadIdx.x;
    const int wave = tid >> 5;
    const int lane = tid & 31;
    const int lm   = lane & 15;      // row (A) / col (B) within 16
    const int lh   = lane >> 4;      // K-pair selector

    const int c4 = tid & 15;         // float4 column for staging
    const int r  = tid >> 4;         // staging row 0..7

    v8f acc[4] = {};
    float4 xr[8], ar[8];

    // Prologue: fetch chunk 0 into registers.
    #pragma unroll
    for (int i = 0; i < 8; ++i) {
        xr[i] = *(const float4*)&x[(size_t)(m0 + r + 8 * i) * D + c4 * 4];
        ar[i] = *(const float4*)&A[(size_t)(r + 8 * i) * D + c4 * 4];
    }

    for (int k0 = 0; k0 < D; k0 += 64) {
        // Stage current chunk registers -> LDS.
        #pragma unroll
        for (int i = 0; i < 8; ++i) {
            *(float4*)&xs[r + 8 * i][c4 * 4] = xr[i];
            *(float4*)&as[r + 8 * i][c4 * 4] = ar[i];
        }
        __syncthreads();

        // Issue next chunk's global loads; results not needed until the
        // ds_store after the next barrier -> overlaps the WMMA loop below.
        const int kn = k0 + 64;
        if (kn < D) {
            #pragma unroll
            for (int i = 0; i < 8; ++i) {
                xr[i] = *(const float4*)&x[(size_t)(m0 + r + 8 * i) * D + kn + c4 * 4];
                ar[i] = *(const float4*)&A[(size_t)(r + 8 * i) * D + kn + c4 * 4];
            }
            // Warm L2 one further chunk ahead (global_prefetch_b8).
            if (kn + 64 < D)
                __builtin_prefetch(&x[(size_t)(m0 + r) * D + kn + 64 + c4 * 4], 0, 0);
        }

        #pragma unroll 4
        for (int kk = 0; kk < 64; kk += 4) {
            const v2f a = *(const v2f*)&xs[wave * 16 + lm][kk + 2 * lh];
            #pragma unroll
            for (int nb = 0; nb < 4; ++nb) {
                const v2f b = *(const v2f*)&as[nb * 16 + lm][kk + 2 * lh];
                acc[nb] = __builtin_amdgcn_wmma_f32_16x16x4_f32(
                    false, a, false, b, (short)0, acc[nb], false, false);
            }
        }
        __syncthreads();
    }

    const int mb = m0 + wave * 16 + lh * 8;   // C layout: vgpr i -> M = i (+8 for hi half)
    #pragma unroll
    for (int nb = 0; nb < 4; ++nb)
        #pragma unroll
        for (int i = 0; i < 8; ++i)
            t[(size_t)(mb + i) * R + nb * 16 + lm] = acc[nb][i];
}

// ---------------------------------------------------------------------------
// Kernel 3: out[M,O] = 2.0 * t[M,64] @ Beff[O,64]^T   (f32 WMMA 16x16x4)
// 256 threads = 8 waves; tile 128(M) x 128(N); K = 64, one LDS stage.
// 4096 blocks -> plenty of waves to hide the single staging phase.
// ---------------------------------------------------------------------------
__global__ __launch_bounds__(256) void gemm2_kernel(
    const float* __restrict__ t, const float* __restrict__ Be,
    float* __restrict__ out, int O)
{
    __shared__ float ts[128][LDSP];
    __shared__ float bs[128][LDSP];

    const int m0   = blockIdx.x * 128;
    const int n0   = blockIdx.y * 128;
    const int tid  = threadIdx.x;
    const int wave = tid >> 5;
    const int lane = tid & 31;
    const int lm   = lane & 15;
    const int lh   = lane >> 4;

    const int c4 = tid & 15;
    const int r  = tid >> 4;         // 0..15, 8 passes cover 128 rows
    #pragma unroll
    for (int i = 0; i < 8; ++i) {
        *(float4*)&ts[r + 16 * i][c4 * 4] =
            *(const float4*)&t[(size_t)(m0 + r + 16 * i) * R + c4 * 4];
        *(float4*)&bs[r + 16 * i][c4 * 4] =
            *(const float4*)&Be[(size_t)(n0 + r + 16 * i) * R + c4 * 4];
    }
    __syncthreads();

    v8f acc[8] = {};
    #pragma unroll
    for (int kk = 0; kk < 64; kk += 4) {
        const v2f a = *(const v2f*)&ts[wave * 16 + lm][kk + 2 * lh];
        #pragma unroll
        for (int nb = 0; nb < 8; ++nb) {
            const v2f b = *(const v2f*)&bs[nb * 16 + lm][kk + 2 * lh];
            acc[nb] = __builtin_amdgcn_wmma_f32_16x16x4_f32(
                false, a, false, b, (short)0, acc[nb], false, false);
        }
    }

    const int mb = m0 + wave * 16 + lh * 8;
    #pragma unroll
    for (int nb = 0; nb < 8; ++nb)
        #pragma unroll
        for (int i = 0; i < 8; ++i)
            out[(size_t)(mb + i) * O + n0 + nb * 16 + lm] = acc[nb][i] * 2.0f;
}

// ---------------------------------------------------------------------------
// Host-side launch. Inputs (setup_inputs order):
// 0:x [B,S,D] 1:lora_A [64,D] 2:lora_B [O,64] 3:hidden_B [O,64]
// 4:W_gate [64,128] 5:b_gate [64] 6:W_tau [64,128] 7:b_tau [64]
// ---------------------------------------------------------------------------
extern "C" void kernel_launch(void* const* d_in, const int* in_sizes, int n_in,
                              void* d_out, int out_size, void* d_ws, size_t ws_size,
                              hipStream_t stream) {
    const float* x      = (const float*)d_in[0];
    const float* lora_A = (const float*)d_in[1];
    const float* lora_B = (const float*)d_in[2];
    const float* hidden = (const float*)d_in[3];
    const float* W_gate = (const float*)d_in[4];
    const float* b_gate = (const float*)d_in[5];
    const float* W_tau  = (const float*)d_in[6];
    const float* b_tau  = (const float*)d_in[7];
    float* out = (float*)d_out;

    const int D = in_sizes[1] / R;          // 4096
    const int O = in_sizes[2] / R;          // 4096
    const int M = in_sizes[0] / D;          // B*S = 16384

    float* Beff = (float*)d_ws;                                          // O*R floats (1 MB)
    float* t    = (float*)((char*)d_ws + (size_t)O * R * sizeof(float)); // M*R floats (4 MB)

    liquid_kernel<<<O / 4, 256, 0, stream>>>(lora_B, hidden, W_gate, b_gate,
                                             W_tau, b_tau, Beff);
    gemm1_kernel<<<M / 64, 128, 0, stream>>>(x, lora_A, t, D);
    dim3 g2(M / 128, O / 128);
    gemm2_kernel<<<g2, 256, 0, stream>>>(t, Beff, out, O);
}